// GraphTransformer_90177133347182
// MI455X (gfx1250) — compile-verified
//
#include <hip/hip_runtime.h>
#include <math.h>

#define N_NODES 50000
#define N_EDGES 800000
#define HEADS 4

typedef __attribute__((ext_vector_type(2))) float v2f;
typedef __attribute__((ext_vector_type(8))) float v8f;

// ---------------------------------------------------------------------------
// WMMA fp32 GEMM:  C[M,N] = A[M,K] @ W[K,N] + bias   (M%16==0, N%64==0, K%4==0)
// One wave32 per 16x64 output tile: 4 accumulators share one A fragment per
// k-step -> 4 back-to-back V_WMMA_F32_16X16X4_F32 on independent accumulators.
// A frag (16x4):  lane holds M=lane&15, K = k0 + 2*(lane>>4) + {0,1}
// B frag (4x16):  lane holds N=lane&15, K = k0 + 2*(lane>>4) + {vgpr 0,1}
// C/D (16x16):    vgpr r -> row = 16*tm + r + 8*(lane>>4), col = lane&15
// ---------------------------------------------------------------------------
__global__ __launch_bounds__(32)
void wmma_gemm_bias(const float* __restrict__ A, const float* __restrict__ W,
                    const float* __restrict__ bias, float* __restrict__ C,
                    int K, int N) {
  const int lane  = threadIdx.x;
  const int tm    = blockIdx.x;
  const int tn    = blockIdx.y;               // 64-wide column tile
  const int m     = tm * 16 + (lane & 15);
  const int n0    = tn * 64 + (lane & 15);
  const int khalf = (lane >> 4) * 2;          // 0 or 2

  const float* arow = A + (size_t)m * K;
  v8f acc0 = {}, acc1 = {}, acc2 = {}, acc3 = {};
#pragma unroll 2
  for (int k0 = 0; k0 < K; k0 += 4) {
    v2f a;
    a.x = arow[k0 + khalf + 0];
    a.y = arow[k0 + khalf + 1];
    const float* w0 = W + (size_t)(k0 + khalf) * N + n0;  // K-row khalf
    const float* w1 = w0 + N;                              // K-row khalf+1
    v2f b0, b1, b2, b3;
    b0.x = w0[0];  b0.y = w1[0];
    b1.x = w0[16]; b1.y = w1[16];
    b2.x = w0[32]; b2.y = w1[32];
    b3.x = w0[48]; b3.y = w1[48];
    acc0 = __builtin_amdgcn_wmma_f32_16x16x4_f32(false, a, false, b0, (short)0, acc0, false, false);
    acc1 = __builtin_amdgcn_wmma_f32_16x16x4_f32(false, a, false, b1, (short)0, acc1, false, false);
    acc2 = __builtin_amdgcn_wmma_f32_16x16x4_f32(false, a, false, b2, (short)0, acc2, false, false);
    acc3 = __builtin_amdgcn_wmma_f32_16x16x4_f32(false, a, false, b3, (short)0, acc3, false, false);
  }
  const int   rbase = tm * 16 + (lane >> 4) * 8;
  const float bn0 = bias[n0], bn1 = bias[n0 + 16], bn2 = bias[n0 + 32], bn3 = bias[n0 + 48];
#pragma unroll
  for (int r = 0; r < 8; ++r) {
    float* crow = C + (size_t)(rbase + r) * N + n0;
    crow[0]  = acc0[r] + bn0;
    crow[16] = acc1[r] + bn1;
    crow[32] = acc2[r] + bn2;
    crow[48] = acc3[r] + bn3;
  }
}

// -------------------- ordered-uint encoding for float atomicMax ------------
__device__ __forceinline__ unsigned f2ord(float f) {
  unsigned u = __float_as_uint(f);
  return (u & 0x80000000u) ? ~u : (u | 0x80000000u);
}
__device__ __forceinline__ float ord2f(unsigned u) {
  return __uint_as_float((u & 0x80000000u) ? (u & 0x7fffffffu) : ~u);
}
#define ORD_NEG_INF 0x007fffffu   // f2ord(-inf)

__global__ void init_nh(unsigned* __restrict__ amax, float* __restrict__ denom, int n) {
  int i = blockIdx.x * blockDim.x + threadIdx.x;
  if (i < n) { amax[i] = ORD_NEG_INF; denom[i] = 0.f; }
}

// alpha[e,h] = (q[dst]·k[src]) / sqrt(C) per head; segment-max into amax[dst,h]
__global__ void edge_alpha_max(const float* __restrict__ q, const float* __restrict__ k,
                               const int* __restrict__ src, const int* __restrict__ dst,
                               float* __restrict__ alpha, unsigned* __restrict__ amax,
                               int F, int C, float inv_sqrt_c) {
  int t = blockIdx.x * blockDim.x + threadIdx.x;
  if (t >= N_EDGES * HEADS) return;
  int e = t >> 2, h = t & 3;
  int s = src[e], d = dst[e];
  const float* qp = q + (size_t)d * F + h * C;
  const float* kp = k + (size_t)s * F + h * C;
  float acc = 0.f;
  for (int c = 0; c < C; ++c) acc += qp[c] * kp[c];
  acc *= inv_sqrt_c;
  alpha[t] = acc;
  atomicMax(&amax[(size_t)d * HEADS + h], f2ord(acc));
}

// decode ordered max; empty segments (-inf) -> 0 per reference
__global__ void amax_fix(unsigned* __restrict__ amax, int n) {
  int i = blockIdx.x * blockDim.x + threadIdx.x;
  if (i < n) {
    unsigned u = amax[i];
    ((float*)amax)[i] = (u == ORD_NEG_INF) ? 0.f : ord2f(u);
  }
}

// ex = exp(alpha - amax[dst]); denom[dst] += ex; alpha <- ex
__global__ void edge_exp_sum(float* __restrict__ alpha, const float* __restrict__ amax,
                             float* __restrict__ denom, const int* __restrict__ dst) {
  int t = blockIdx.x * blockDim.x + threadIdx.x;
  if (t >= N_EDGES * HEADS) return;
  int e = t >> 2, h = t & 3;
  int d = dst[e];
  float ex = expf(alpha[t] - amax[(size_t)d * HEADS + h]);
  alpha[t] = ex;
  atomicAdd(&denom[(size_t)d * HEADS + h], ex);
}

// out[dst] += (ex/denom[dst]) * v[src]    (out pre-filled with x@Ws + bs)
__global__ void edge_aggregate(const float* __restrict__ alpha, const float* __restrict__ denom,
                               const float* __restrict__ v,
                               const int* __restrict__ src, const int* __restrict__ dst,
                               float* __restrict__ out, int F, int C) {
  int t = blockIdx.x * blockDim.x + threadIdx.x;
  if (t >= N_EDGES * HEADS) return;
  int e = t >> 2, h = t & 3;
  int s = src[e], d = dst[e];
  float w = alpha[t] / (denom[(size_t)d * HEADS + h] + 1e-16f);
  const float* vp = v + (size_t)s * F + h * C;
  float* op = out + (size_t)d * F + h * C;
  for (int c = 0; c < C; ++c) atomicAdd(&op[c], w * vp[c]);
}

// LayerNorm + ELU (+optional residual added AFTER the ELU); one wave per row.
__global__ __launch_bounds__(256)
void ln_elu(const float* __restrict__ x, const float* __restrict__ g, const float* __restrict__ b,
            const float* __restrict__ res, float* __restrict__ out, int F) {
  int wave = (blockIdx.x * blockDim.x + threadIdx.x) >> 5;
  int lane = threadIdx.x & 31;
  if (wave >= N_NODES) return;
  const float* row = x + (size_t)wave * F;
  const int per = F >> 5;                 // 4 (F=128) or 8 (F=256)
  float vals[8];
  float sum = 0.f;
  for (int i = 0; i < per; ++i) { vals[i] = row[lane + i * 32]; sum += vals[i]; }
  for (int off = 16; off; off >>= 1) sum += __shfl_xor(sum, off, 32);
  float mu = sum / (float)F;
  float vs = 0.f;
  for (int i = 0; i < per; ++i) { float d = vals[i] - mu; vs += d * d; }
  for (int off = 16; off; off >>= 1) vs += __shfl_xor(vs, off, 32);
  float rstd = rsqrtf(vs / (float)F + 1e-5f);
  float* orow = out + (size_t)wave * F;
  for (int i = 0; i < per; ++i) {
    int c = lane + i * 32;
    float y = (vals[i] - mu) * rstd * g[c] + b[c];
    y = (y > 0.f) ? y : (expf(y) - 1.f);          // ELU
    if (res) y += res[(size_t)wave * F + c];      // h4_in = h3 + h1
    orow[c] = y;
  }
}

// Layer-4 epilogue: out_conv = O4; out_norm = LN(O4); logits = ELU(out_norm)@Wc+bc
__global__ __launch_bounds__(256)
void final_head(const float* __restrict__ o4, const float* __restrict__ g, const float* __restrict__ b,
                const float* __restrict__ Wc, const float* __restrict__ bc,
                float* __restrict__ logits, float* __restrict__ out_conv,
                float* __restrict__ out_norm) {
  int wave = (blockIdx.x * blockDim.x + threadIdx.x) >> 5;
  int lane = threadIdx.x & 31;
  if (wave >= N_NODES) return;
  const float* row = o4 + (size_t)wave * 64;
  float v0 = row[lane], v1 = row[lane + 32];
  float sum = v0 + v1;
  for (int off = 16; off; off >>= 1) sum += __shfl_xor(sum, off, 32);
  float mu = sum * (1.f / 64.f);
  float d0 = v0 - mu, d1 = v1 - mu;
  float vs = d0 * d0 + d1 * d1;
  for (int off = 16; off; off >>= 1) vs += __shfl_xor(vs, off, 32);
  float rstd = rsqrtf(vs * (1.f / 64.f) + 1e-5f);
  float n0 = d0 * rstd * g[lane] + b[lane];
  float n1 = d1 * rstd * g[lane + 32] + b[lane + 32];
  out_conv[(size_t)wave * 64 + lane]      = v0;
  out_conv[(size_t)wave * 64 + lane + 32] = v1;
  out_norm[(size_t)wave * 64 + lane]      = n0;
  out_norm[(size_t)wave * 64 + lane + 32] = n1;
  float h0 = (n0 > 0.f) ? n0 : (expf(n0) - 1.f);
  float h1 = (n1 > 0.f) ? n1 : (expf(n1) - 1.f);
  float l0 = h0 * Wc[lane * 2 + 0] + h1 * Wc[(lane + 32) * 2 + 0];
  float l1 = h0 * Wc[lane * 2 + 1] + h1 * Wc[(lane + 32) * 2 + 1];
  for (int off = 16; off; off >>= 1) { l0 += __shfl_xor(l0, off, 32); l1 += __shfl_xor(l1, off, 32); }
  if (lane == 0) {
    logits[(size_t)wave * 2 + 0] = l0 + bc[0];
    logits[(size_t)wave * 2 + 1] = l1 + bc[1];
  }
}

// ---------------------------------------------------------------------------
static void run_layer(const float* xin, int fin, int fout,
                      const float* Wq, const float* bq, const float* Wk, const float* bk,
                      const float* Wv, const float* bv, const float* Ws, const float* bs,
                      float* q, float* k, float* v, float* out,
                      float* alpha, float* amax, float* denom,
                      const int* src, const int* dst, hipStream_t stream) {
  dim3 gg(N_NODES / 16, fout / 64);
  wmma_gemm_bias<<<gg, 32, 0, stream>>>(xin, Wq, bq, q,   fin, fout);
  wmma_gemm_bias<<<gg, 32, 0, stream>>>(xin, Wk, bk, k,   fin, fout);
  wmma_gemm_bias<<<gg, 32, 0, stream>>>(xin, Wv, bv, v,   fin, fout);
  wmma_gemm_bias<<<gg, 32, 0, stream>>>(xin, Ws, bs, out, fin, fout);

  const int C = fout / HEADS;
  const int nh = N_NODES * HEADS;
  const int eh = N_EDGES * HEADS;
  init_nh<<<(nh + 255) / 256, 256, 0, stream>>>((unsigned*)amax, denom, nh);
  edge_alpha_max<<<(eh + 255) / 256, 256, 0, stream>>>(q, k, src, dst, alpha, (unsigned*)amax,
                                                       fout, C, rsqrtf((float)C));
  amax_fix<<<(nh + 255) / 256, 256, 0, stream>>>((unsigned*)amax, nh);
  edge_exp_sum<<<(eh + 255) / 256, 256, 0, stream>>>(alpha, amax, denom, dst);
  edge_aggregate<<<(eh + 255) / 256, 256, 0, stream>>>(alpha, denom, v, src, dst, out, fout, C);
}

extern "C" void kernel_launch(void* const* d_in, const int* in_sizes, int n_in,
                              void* d_out, int out_size, void* d_ws, size_t ws_size,
                              hipStream_t stream) {
  (void)in_sizes; (void)n_in; (void)out_size; (void)ws_size;

  // ---- inputs, in setup_inputs() dict insertion order ----
  const float* x   = (const float*)d_in[0];
  const int*   ei  = (const int*)d_in[1];
  const int*   src = ei;
  const int*   dst = ei + N_EDGES;
  // conv params: base index 2 + 8*layer, order Wq,bq,Wk,bk,Wv,bv,Ws,bs
  const float* cp[4][8];
  for (int l = 0; l < 4; ++l)
    for (int j = 0; j < 8; ++j)
      cp[l][j] = (const float*)d_in[2 + 8 * l + j];
  const float* ln_g[4] = { (const float*)d_in[34], (const float*)d_in[36],
                           (const float*)d_in[38], (const float*)d_in[40] };
  const float* ln_b[4] = { (const float*)d_in[35], (const float*)d_in[37],
                           (const float*)d_in[39], (const float*)d_in[41] };
  const float* Wc = (const float*)d_in[42];
  const float* bc = (const float*)d_in[43];

  // ---- workspace carve-up (floats) ----
  float* W = (float*)d_ws;
  size_t o = 0;
  const size_t NMAX = (size_t)N_NODES * 256;
  float* q     = W + o; o += NMAX;
  float* k     = W + o; o += NMAX;
  float* v     = W + o; o += NMAX;
  float* out   = W + o; o += NMAX;
  float* bufB  = W + o; o += NMAX;                       // h2, then h4_in
  float* h1    = W + o; o += (size_t)N_NODES * 128;      // persists for residual
  float* alpha = W + o; o += (size_t)N_EDGES * HEADS;
  float* amax  = W + o; o += (size_t)N_NODES * HEADS;
  float* denom = W + o; o += (size_t)N_NODES * HEADS;

  const int lnBlocks = (N_NODES * 32 + 255) / 256;       // one wave per row

  // ---- layer 1: 256 -> 128 ----
  run_layer(x, 256, 128, cp[0][0], cp[0][1], cp[0][2], cp[0][3], cp[0][4], cp[0][5],
            cp[0][6], cp[0][7], q, k, v, out, alpha, amax, denom, src, dst, stream);
  ln_elu<<<lnBlocks, 256, 0, stream>>>(out, ln_g[0], ln_b[0], nullptr, h1, 128);

  // ---- layer 2: 128 -> 256 ----
  run_layer(h1, 128, 256, cp[1][0], cp[1][1], cp[1][2], cp[1][3], cp[1][4], cp[1][5],
            cp[1][6], cp[1][7], q, k, v, out, alpha, amax, denom, src, dst, stream);
  ln_elu<<<lnBlocks, 256, 0, stream>>>(out, ln_g[1], ln_b[1], nullptr, bufB, 256);

  // ---- layer 3: 256 -> 128 ; h4_in = elu(ln(out)) + h1 ----
  run_layer(bufB, 256, 128, cp[2][0], cp[2][1], cp[2][2], cp[2][3], cp[2][4], cp[2][5],
            cp[2][6], cp[2][7], q, k, v, out, alpha, amax, denom, src, dst, stream);
  ln_elu<<<lnBlocks, 256, 0, stream>>>(out, ln_g[2], ln_b[2], h1, bufB, 128);

  // ---- layer 4: 128 -> 64 ----
  run_layer(bufB, 128, 64, cp[3][0], cp[3][1], cp[3][2], cp[3][3], cp[3][4], cp[3][5],
            cp[3][6], cp[3][7], q, k, v, out, alpha, amax, denom, src, dst, stream);

  // ---- epilogue: logits | out_conv | out_norm (concatenated in return order) ----
  float* d_logits   = (float*)d_out;
  float* d_out_conv = d_logits + (size_t)N_NODES * 2;
  float* d_out_norm = d_out_conv + (size_t)N_NODES * 64;
  final_head<<<lnBlocks, 256, 0, stream>>>(out, ln_g[3], ln_b[3], Wc, bc,
                                           d_logits, d_out_conv, d_out_norm);
}